// Model_77378130805194
// MI455X (gfx1250) — compile-verified
//
#include <hip/hip_runtime.h>
#include <hip/hip_bf16.h>
#include <math.h>

#define NTIME  128
#define NBATCH 1024
#define NSAMP  (NTIME * NBATCH)
#define WAVES  4

typedef float v2f __attribute__((ext_vector_type(2)));
typedef float v8f __attribute__((ext_vector_type(8)));

__device__ __forceinline__ v8f wmma4(v2f a, v2f b, v8f c) {
  // D = A(16x4) x B(4x16) + C(16x16), f32, wave32
  return __builtin_amdgcn_wmma_f32_16x16x4_f32(false, a, false, b, (short)0, c, false, false);
}

// One forward layer: hb(new 64x16) = relu(W(64xKdim) @ hb(old Kdimx16) + bias),
// recording relu' bits into mword[0..1] per sample column.
__device__ __forceinline__ void fwd_layer(const float* W, int Kld, int Kdim,
                                          const float* bias, float* hb,
                                          unsigned* mword, int l16, int half)
{
  v8f t[4];
#pragma unroll
  for (int rt = 0; rt < 4; ++rt) {
    v8f acc = {};
    for (int k0 = 0; k0 < Kdim; k0 += 4) {
      int ka = k0 + 2 * half;
      v2f a, b;
      a[0] = W[(rt * 16 + l16) * Kld + ka];
      a[1] = W[(rt * 16 + l16) * Kld + ka + 1];
      b[0] = hb[ka * 16 + l16];
      b[1] = hb[(ka + 1) * 16 + l16];
      acc = wmma4(a, b, acc);
    }
    t[rt] = acc;
  }
  // all reads of hb done; now bias + relu + mask, write back
#pragma unroll
  for (int rt = 0; rt < 4; ++rt) {
    unsigned bb = 0u;
#pragma unroll
    for (int v = 0; v < 8; ++v) {
      int r = rt * 16 + v + 8 * half;       // hidden-unit row
      float pre = t[rt][v] + bias[r];
      unsigned m = pre > 0.f ? 1u : 0u;
      bb |= m << v;
      hb[r * 16 + l16] = m ? pre : 0.f;
    }
    atomicOr(mword + (rt >> 1), bb << ((rt * 16 + 8 * half) & 31));
  }
}

__global__ __launch_bounds__(WAVES * 32)
void visco_jac_kernel(const float* __restrict__ y, const float* __restrict__ erate,
                      const float* __restrict__ Tt, const float* __restrict__ Ep,
                      const float* __restrict__ np_, const float* __restrict__ w_start,
                      const float* __restrict__ b_start, const float* __restrict__ w_int,
                      const float* __restrict__ b_int, const float* __restrict__ w_end,
                      const float* __restrict__ b_end, float* __restrict__ out)
{
  __shared__ float    wsL[64][16];        // w_start padded cols 10..15 = 0
  __shared__ float    wiL[2][64][64];
  __shared__ float    weL[16][64];        // w_end padded rows 8..15 = 0
  __shared__ float    bsL[64];
  __shared__ float    biL[2][64];
  __shared__ float    beL[16];
  __shared__ float    HbL[WAVES][64 * 16]; // activations (64x16) / adjoint M (16x64)
  __shared__ unsigned mkb[WAVES][3][16][2]; // relu' bits: [layer][sample][word]
  __shared__ float    scalb[WAVES][16][2];  // {coef, add0} per sample

  const int tid  = threadIdx.x;
  const int wv   = tid >> 5;
  const int lane = tid & 31;
  const int half = lane >> 4;
  const int l16  = lane & 15;
  const int base = (blockIdx.x * WAVES + wv) * 16;   // first sample of this wave

  // ---- cooperative weight load (block-wide) ----
  for (int i = tid; i < 64 * 16; i += WAVES * 32) {
    int r = i >> 4, c = i & 15;
    wsL[r][c] = (c < 10) ? w_start[r * 10 + c] : 0.f;
  }
  for (int i = tid; i < 2 * 64 * 64; i += WAVES * 32) ((float*)wiL)[i] = w_int[i];
  for (int i = tid; i < 16 * 64; i += WAVES * 32) {
    int r = i >> 6, c = i & 63;
    weL[r][c] = (r < 8) ? w_end[r * 64 + c] : 0.f;
  }
  for (int i = tid; i < 64; i += WAVES * 32) bsL[i] = b_start[i];
  for (int i = tid; i < 2 * 64; i += WAVES * 32) ((float*)biL)[i] = b_int[i];
  for (int i = tid; i < 16; i += WAVES * 32) beL[i] = (i < 8) ? b_end[i] : 0.f;
  __syncthreads();

  float* hb = HbL[wv];

  // ---- wave-local: zero mask bits, load X (16 rows x 16 samples) into hb ----
  for (int i = lane; i < 3 * 16 * 2; i += 32) ((unsigned*)mkb[wv])[i] = 0u;
  for (int idx = lane; idx < 256; idx += 32) {
    int k = idx >> 4, nn = idx & 15;
    int gs = base + nn;
    float v = 0.f;
    if (k < 8)       v = y[(size_t)gs * 8 + k];
    else if (k == 8) v = erate[gs];
    else if (k == 9) v = Tt[gs];
    hb[k * 16 + nn] = v;
  }

  const float eE = expf(Ep[0]);
  const float pn = expf(np_[0]);

  // ---- forward: 3 hidden layers + output ----
  fwd_layer(&wsL[0][0],    16, 16, bsL,    hb, &mkb[wv][0][l16][0], l16, half);
  fwd_layer(&wiL[0][0][0], 64, 64, biL[0], hb, &mkb[wv][1][l16][0], l16, half);
  fwd_layer(&wiL[1][0][0], 64, 64, biL[1], hb, &mkb[wv][2][l16][0], l16, half);

  {
    v8f o = {};
    for (int k0 = 0; k0 < 64; k0 += 4) {
      int ka = k0 + 2 * half;
      v2f a, b;
      a[0] = weL[l16][ka];
      a[1] = weL[l16][ka + 1];
      b[0] = hb[ka * 16 + l16];
      b[1] = hb[(ka + 1) * 16 + l16];
      o = wmma4(a, b, o);
    }
    // lanes 0..15 hold all 8 output rows for sample l16
    if (half == 0) {
      int gs = base + l16;
      float s0 = o[0] + beL[0];
      float y0 = y[(size_t)gs * 8];
      float er = erate[gs];
      float sg  = (float)((y0 > 0.f) - (y0 < 0.f));
      float u   = fabsf(y0) * (1.f + s0 * 1e-5f);
      float g   = powf(u, pn);
      float mac = 0.5f * (g + fabsf(g));
      float gsg = (float)((g > 0.f) - (g < 0.f));
      float macd = 0.5f * (1.f + gsg);              // macaulay'
      float dgdu = pn * powf(u, pn - 1.f);
      float coef = -eE * sg * macd * dgdu * fabsf(y0) * 1e-5f;    // * ds0/dx_j
      float add0 = -eE * macd * dgdu * (sg * sg) * (1.f + s0 * 1e-5f); // direct d|y0|
      scalb[wv][l16][0] = coef;
      scalb[wv][l16][1] = add0;
      out[(size_t)gs * 8 + 0] = eE * (er - mac * sg);
#pragma unroll
      for (int i = 1; i < 8; ++i)
        out[(size_t)gs * 8 + i] = (o[i] + beL[i]) * 1e-4f;
    }
  }

  // ---- reverse-mode Jacobian: 2 samples per 16x64 adjoint block ----
  float* J1 = out + (size_t)NSAMP * 8;
  float* J2 = J1 + (size_t)NSAMP * 64;
  float* J3 = J2 + (size_t)NSAMP * 8;

  for (int p = 0; p < 8; ++p) {
    // init M[16][64] = [We ⊙ mask2(sA); We ⊙ mask2(sB)]  into hb
    for (int idx = lane; idx < 16 * 64; idx += 32) {
      int m = idx >> 6, col = idx & 63;
      int ss = 2 * p + (m >> 3);
      unsigned mw = mkb[wv][2][ss][col >> 5];
      float mk = (float)((mw >> (col & 31)) & 1u);
      hb[idx] = weL[m & 7][col] * mk;
    }
    const int s = 2 * p + half;     // this lane's sample (C/D-tile row half)

    for (int l = 1; l >= 0; --l) {  // M <- (M @ Wi[l]) ⊙ mask[l]
      v8f Mt[4];
#pragma unroll
      for (int ct = 0; ct < 4; ++ct) {
        v8f acc = {};
        for (int k0 = 0; k0 < 64; k0 += 4) {
          int ka = k0 + 2 * half;
          v2f a, b;
          a[0] = hb[l16 * 64 + ka];
          a[1] = hb[l16 * 64 + ka + 1];
          b[0] = wiL[l][ka][ct * 16 + l16];
          b[1] = wiL[l][ka + 1][ct * 16 + l16];
          acc = wmma4(a, b, acc);
        }
        Mt[ct] = acc;
      }
      unsigned mw0 = mkb[wv][l][s][0], mw1 = mkb[wv][l][s][1];
#pragma unroll
      for (int ct = 0; ct < 4; ++ct) {
        int col = ct * 16 + l16;
        unsigned mw = (col < 32) ? mw0 : mw1;
        float mk = (float)((mw >> (col & 31)) & 1u);
#pragma unroll
        for (int v = 0; v < 8; ++v)
          hb[(v + 8 * half) * 64 + col] = Mt[ct][v] * mk;
      }
    }

    // J(16x16) = M(16x64) @ Ws(64x16-padded)
    v8f Jt = {};
    for (int k0 = 0; k0 < 64; k0 += 4) {
      int ka = k0 + 2 * half;
      v2f a, b;
      a[0] = hb[l16 * 64 + ka];
      a[1] = hb[l16 * 64 + ka + 1];
      b[0] = wsL[ka][l16];
      b[1] = wsL[ka + 1][l16];
      Jt = wmma4(a, b, Jt);
    }

    // emit: lane holds J[i=v][j=l16] for sample s
    int j = l16;
    int gs = base + s;
    float coef = scalb[wv][s][0];
    float add0 = scalb[wv][s][1];
    if (j < 8) {
      J1[(size_t)gs * 64 + j] = coef * Jt[0] + ((j == 0) ? add0 : 0.f);
#pragma unroll
      for (int i = 1; i < 8; ++i) J1[(size_t)gs * 64 + i * 8 + j] = Jt[i] * 1e-4f;
    } else if (j == 8) {
      J2[(size_t)gs * 8 + 0] = eE + coef * Jt[0];
#pragma unroll
      for (int i = 1; i < 8; ++i) J2[(size_t)gs * 8 + i] = Jt[i] * 1e-4f;
    } else if (j == 9) {
      J3[(size_t)gs * 8 + 0] = coef * Jt[0];
#pragma unroll
      for (int i = 1; i < 8; ++i) J3[(size_t)gs * 8 + i] = Jt[i] * 1e-4f;
    }
  }
}

extern "C" void kernel_launch(void* const* d_in, const int* in_sizes, int n_in,
                              void* d_out, int out_size, void* d_ws, size_t ws_size,
                              hipStream_t stream) {
  (void)in_sizes; (void)n_in; (void)d_ws; (void)ws_size; (void)out_size;
  // dict order: t, y, erate, T, E, n, w_start, b_start, w_int, b_int, w_end, b_end
  const float* y  = (const float*)d_in[1];
  const float* er = (const float*)d_in[2];
  const float* T  = (const float*)d_in[3];
  const float* E  = (const float*)d_in[4];
  const float* n  = (const float*)d_in[5];
  const float* ws = (const float*)d_in[6];
  const float* bs = (const float*)d_in[7];
  const float* wi = (const float*)d_in[8];
  const float* bi = (const float*)d_in[9];
  const float* we = (const float*)d_in[10];
  const float* be = (const float*)d_in[11];
  float* out = (float*)d_out;

  dim3 grid(NSAMP / (WAVES * 16));   // 2048 blocks
  dim3 block(WAVES * 32);            // 128 threads = 4 waves
  hipLaunchKernelGGL(visco_jac_kernel, grid, block, 0, stream,
                     y, er, T, E, n, ws, bs, wi, bi, we, be, out);
}